// MinibatchDiscrimination_41540923686995
// MI455X (gfx1250) — compile-verified
//
#include <hip/hip_runtime.h>

typedef float v2f __attribute__((ext_vector_type(2)));
typedef float v8f __attribute__((ext_vector_type(8)));

#define B_SZ    512
#define IN_SZ   512
#define OUT_SZ  64
#define KD_SZ   8
#define WCOLS   (OUT_SZ * KD_SZ)     // 512
#define OUTCOLS (IN_SZ + OUT_SZ)     // 576
#define TI      16                   // i-rows per block in pairwise kernel

// ---------------------------------------------------------------------------
// Kernel 1: M = x @ W   (512x512 @ 512x512, f32) via V_WMMA_F32_16X16X4_F32.
// M is written into d_out[:, 0:512] (row stride OUTCOLS) as scratch.
// One wave computes one 16x16 C tile; K-loop in steps of 4.
// A 16x4 layout: lanes 0-15 -> M=lane, VGPR{0,1}=K{0,1}; lanes 16-31 -> K{2,3}
// B 4x16 layout: lanes 0-15 -> N=lane, VGPR{0,1}=K{0,1}; lanes 16-31 -> K{2,3}
// C/D layout:    VGPR v, lanes 0-15 -> M=v, lanes 16-31 -> M=v+8, N=lane&15
// ---------------------------------------------------------------------------
__global__ __launch_bounds__(256) void mbd_gemm_wmma(
    const float* __restrict__ x, const float* __restrict__ W,
    float* __restrict__ out)
{
    const int tid  = threadIdx.x;
    const int lane = tid & 31;
    const int wid  = (blockIdx.x << 3) + (tid >> 5);  // 0..1023 tile id
    const int tm   = wid >> 5;                        // 0..31
    const int tn   = wid & 31;                        // 0..31

    const int half = lane >> 4;                       // 0 or 1
    const int l16  = lane & 15;
    const int arow = tm * 16 + l16;
    const int bcol = tn * 16 + l16;

    v8f c = {};
    for (int k = 0; k < IN_SZ; k += 4) {
        const int ka = k + 2 * half;
        v2f a;
        a.x = x[arow * IN_SZ + ka];
        a.y = x[arow * IN_SZ + ka + 1];
        v2f b;
        b.x = W[(ka    ) * WCOLS + bcol];
        b.y = W[(ka + 1) * WCOLS + bcol];
        // (neg_a, A, neg_b, B, c_mod, C, reuse_a, reuse_b)
        c = __builtin_amdgcn_wmma_f32_16x16x4_f32(
                false, a, false, b, (short)0, c, false, false);
    }

    #pragma unroll
    for (int v = 0; v < 8; ++v) {
        const int row = tm * 16 + v + 8 * half;
        out[row * OUTCOLS + bcol] = c[v];
    }
}

// ---------------------------------------------------------------------------
// Kernel 2: out[i, 512+o] = sum_j exp(-sum_k |M[i,o,k]-M[j,o,k]|)
// Block handles TI=16 i-rows; 256 threads = 64 o-values x 4 i-groups.
// j-rows staged through LDS in chunks of TI (16*512*4B = 32 KB).
// Reads M from out[:, 0:512], writes out[:, 512:576] (disjoint regions).
// ---------------------------------------------------------------------------
__global__ __launch_bounds__(256) void mbd_pairwise(
    const float* Mbuf, float* out)
{
    __shared__ float sj[TI * WCOLS];

    const int tid   = threadIdx.x;
    const int o     = tid & 63;     // output feature 0..63
    const int ig    = tid >> 6;     // 0..3
    const int ibase = blockIdx.x * TI;

    // Cache this thread's 4 i-rows (8 kernel dims each) in registers.
    float mi[4][KD_SZ];
    #pragma unroll
    for (int r = 0; r < 4; ++r) {
        const int i = ibase + ig * 4 + r;
        #pragma unroll
        for (int k = 0; k < KD_SZ; ++k)
            mi[r][k] = Mbuf[i * OUTCOLS + o * KD_SZ + k];
    }

    float acc[4] = {0.f, 0.f, 0.f, 0.f};

    for (int jt = 0; jt < B_SZ; jt += TI) {
        __syncthreads();
        // Cooperative load of TI j-rows (16 x 512 floats) as float4.
        for (int u = tid; u < TI * (WCOLS / 4); u += 256) {
            const int jr = u >> 7;          // row within chunk (512/4 = 128)
            const int cc = u & 127;         // float4 column
            const float4* src = (const float4*)(Mbuf + (jt + jr) * OUTCOLS);
            ((float4*)sj)[jr * (WCOLS / 4) + cc] = src[cc];
        }
        __syncthreads();

        for (int jj = 0; jj < TI; ++jj) {
            float mj[KD_SZ];
            #pragma unroll
            for (int k = 0; k < KD_SZ; ++k)
                mj[k] = sj[jj * WCOLS + o * KD_SZ + k];
            #pragma unroll
            for (int r = 0; r < 4; ++r) {
                float d = 0.f;
                #pragma unroll
                for (int k = 0; k < KD_SZ; ++k)
                    d += fabsf(mi[r][k] - mj[k]);
                acc[r] += __expf(-d);       // d >= 0, no overflow
            }
        }
    }

    #pragma unroll
    for (int r = 0; r < 4; ++r) {
        const int i = ibase + ig * 4 + r;
        out[i * OUTCOLS + IN_SZ + o] = acc[r];
    }
}

// ---------------------------------------------------------------------------
// Kernel 3: out[:, 0:512] = x  (overwrites the M scratch region last)
// ---------------------------------------------------------------------------
__global__ __launch_bounds__(256) void mbd_copy_x(
    const float* __restrict__ x, float* __restrict__ out)
{
    const int idx = blockIdx.x * 256 + threadIdx.x;   // 0 .. 65535 float4 slots
    const int i   = idx >> 7;                         // row (128 float4 per row)
    const int c   = idx & 127;
    ((float4*)(out + i * OUTCOLS))[c] = ((const float4*)(x + i * IN_SZ))[c];
}

extern "C" void kernel_launch(void* const* d_in, const int* in_sizes, int n_in,
                              void* d_out, int out_size, void* d_ws, size_t ws_size,
                              hipStream_t stream) {
    (void)in_sizes; (void)n_in; (void)out_size; (void)d_ws; (void)ws_size;
    const float* x = (const float*)d_in[0];
    const float* T = (const float*)d_in[1];   // [512, 64, 8] contiguous == [512, 512]
    float* out = (float*)d_out;               // [512, 576]

    mbd_gemm_wmma<<<128, 256, 0, stream>>>(x, T, out);
    mbd_pairwise<<<B_SZ / TI, 256, 0, stream>>>(out, out);
    mbd_copy_x<<<(B_SZ * IN_SZ / 4) / 256, 256, 0, stream>>>(x, out);
}